// DBMGNet_47871705481780
// MI455X (gfx1250) — compile-verified
//
#include <hip/hip_runtime.h>
#include <math.h>

// ---------------- constants ----------------
static constexpr int Bn   = 32;    // batch
static constexpr int Cch  = 200;   // input channels
static constexpr int Himg = 16;
static constexpr int HWn  = 256;   // Himg*Himg
static constexpr int Emb  = 128;
static constexpr int NCls = 16;

typedef float v2f __attribute__((ext_vector_type(2)));
typedef float v8f __attribute__((ext_vector_type(8)));

// ---------------- activation helper ----------------
__device__ __forceinline__ float act_apply(float x, int act) {
  switch (act) {
    case 1: return x > 0.f ? x : 0.f;                               // relu
    case 2: return 1.f / (1.f + __expf(-x));                        // sigmoid
    case 3: return x / (1.f + __expf(-x));                          // silu
    case 4: return 0.5f * x * (1.f + erff(x * 0.70710678118f));     // gelu (exact)
    case 5: return x > 20.f ? x : log1pf(__expf(x));                // softplus
    default: return x;
  }
}

// ---------------- WMMA fp32 GEMM: C[b] = act(alpha * A[b]@B[b] + bias) ----------------
// One wave32 computes a 32x32 C tile with 4 accumulators (A/B fragments reused 2x).
// K must be a multiple of 4 (true for every GEMM in this net).  Row/col guards are
// loop-invariant: clamped addresses + value selects, no exec-divergent branches, so
// the inner loop is 8 loads + 4 v_wmma_f32_16x16x4_f32 + pointer bumps.
// TA: A[m][k] = Ap[k*lda+m]   TB: B[k][n] = Bp[n*ldb+k]
template <int TA, int TB>
__global__ void gemm_wmma_f32(const float* __restrict__ A, const float* __restrict__ Bp,
                              const float* __restrict__ bias, float* __restrict__ C,
                              int M, int N, int K, int lda, int ldb, int ldc,
                              long sA, long sB, long sC, float alpha, int act) {
  const int lane = threadIdx.x & 31;
  const int tn = blockIdx.x * 32;
  const int tm = blockIdx.y * 32;
  const long b = blockIdx.z;
  const float* Ab = A + b * sA;
  const float* Bb = Bp + b * sB;
  float* Cb = C + b * sC;

  const int l15  = lane & 15;
  const int half = lane >> 4;     // selects K pair {k,k+1} vs {k+2,k+3}
  const int kh   = half * 2;

  const int r0 = tm + l15,      r1 = tm + 16 + l15;
  const bool r0v = r0 < M,      r1v = r1 < M;
  const int r0c = r0v ? r0 : M - 1, r1c = r1v ? r1 : M - 1;
  const int c0 = tn + l15,      c1 = tn + 16 + l15;
  const bool c0v = c0 < N,      c1v = c1 < N;
  const int c0c = c0v ? c0 : N - 1, c1c = c1v ? c1 : N - 1;

  const float* a0p; const float* a1p; long aInc; int aS;
  if (TA == 0) {
    a0p = Ab + (long)r0c * lda + kh; a1p = Ab + (long)r1c * lda + kh; aInc = 4;            aS = 1;
  } else {
    a0p = Ab + (long)kh * lda + r0c; a1p = Ab + (long)kh * lda + r1c; aInc = 4 * (long)lda; aS = lda;
  }
  const float* b0p; const float* b1p; long bInc; int bS;
  if (TB == 1) {
    b0p = Bb + (long)c0c * ldb + kh; b1p = Bb + (long)c1c * ldb + kh; bInc = 4;            bS = 1;
  } else {
    b0p = Bb + (long)kh * ldb + c0c; b1p = Bb + (long)kh * ldb + c1c; bInc = 4 * (long)ldb; bS = ldb;
  }

  v8f acc00 = {}, acc01 = {}, acc10 = {}, acc11 = {};
  for (int k0 = 0; k0 < K; k0 += 4) {
    v2f a0, a1, b0, b1;
    a0[0] = a0p[0]; a0[1] = a0p[aS];
    a1[0] = a1p[0]; a1[1] = a1p[aS];
    b0[0] = b0p[0]; b0[1] = b0p[bS];
    b1[0] = b1p[0]; b1[1] = b1p[bS];
    if (!r0v) { a0[0] = 0.f; a0[1] = 0.f; }
    if (!r1v) { a1[0] = 0.f; a1[1] = 0.f; }
    if (!c0v) { b0[0] = 0.f; b0[1] = 0.f; }
    if (!c1v) { b1[0] = 0.f; b1[1] = 0.f; }
    acc00 = __builtin_amdgcn_wmma_f32_16x16x4_f32(false, a0, false, b0, (short)0, acc00, false, false);
    acc01 = __builtin_amdgcn_wmma_f32_16x16x4_f32(false, a0, false, b1, (short)0, acc01, false, false);
    acc10 = __builtin_amdgcn_wmma_f32_16x16x4_f32(false, a1, false, b0, (short)0, acc10, false, false);
    acc11 = __builtin_amdgcn_wmma_f32_16x16x4_f32(false, a1, false, b1, (short)0, acc11, false, false);
    a0p += aInc; a1p += aInc; b0p += bInc; b1p += bInc;
  }

  auto store_tile = [&](const v8f& acc, int baseM, int colIdx, bool colValid) {
    if (!colValid) return;
    const float bs = bias ? bias[colIdx] : 0.f;
#pragma unroll
    for (int r = 0; r < 8; ++r) {
      const int row = baseM + r + half * 8;
      if (row < M) {
        float v = acc[r] * alpha + bs;
        Cb[(long)row * ldc + colIdx] = act_apply(v, act);
      }
    }
  };
  store_tile(acc00, tm,      c0, c0v);
  store_tile(acc01, tm,      c1, c1v);
  store_tile(acc10, tm + 16, c0, c0v);
  store_tile(acc11, tm + 16, c1, c1v);
}

// ---------------- elementwise / structural kernels ----------------
__global__ void bn_relu_k(const float* __restrict__ x, const float* __restrict__ g,
                          const float* __restrict__ bb, const float* __restrict__ m,
                          const float* __restrict__ v, float* __restrict__ out,
                          long total, int Ch, int HWc) {
  long i = blockIdx.x * (long)blockDim.x + threadIdx.x;
  if (i >= total) return;
  int c = (int)((i / HWc) % Ch);
  float val = (x[i] - m[c]) * rsqrtf(v[c] + 1e-5f) * g[c] + bb[c];
  out[i] = fmaxf(val, 0.f);
}

__global__ void dwconv3x3_k(const float* __restrict__ x, const float* __restrict__ w,
                            float* __restrict__ out, int Bc, int Ch, int H) {
  long total = (long)Bc * Ch * H * H;
  long i = blockIdx.x * (long)blockDim.x + threadIdx.x;
  if (i >= total) return;
  int xx = (int)(i % H); long t = i / H;
  int yy = (int)(t % H); t /= H;
  int c  = (int)(t % Ch); int b = (int)(t / Ch);
  const float* xb = x + ((long)b * Ch + c) * H * H;
  const float* wc = w + c * 9;
  float s = 0.f;
  for (int dy = 0; dy < 3; ++dy) {
    int ys = yy + dy - 1;
    if (ys < 0 || ys >= H) continue;
    for (int dx = 0; dx < 3; ++dx) {
      int xs = xx + dx - 1;
      if (xs < 0 || xs >= H) continue;
      s += wc[dy * 3 + dx] * xb[ys * H + xs];
    }
  }
  out[i] = s;
}

__global__ void add2_k(float* __restrict__ out, const float* __restrict__ a,
                       const float* __restrict__ b, long total) {
  long i = blockIdx.x * (long)blockDim.x + threadIdx.x;
  if (i >= total) return;
  out[i] = a[i] + b[i];
}

__global__ void rope_pos_k(const float* __restrict__ xe, const float* __restrict__ pos,
                           float* __restrict__ out, int Bc, int L, int E) {
  long total = (long)Bc * L * E;
  long i = blockIdx.x * (long)blockDim.x + threadIdx.x;
  if (i >= total) return;
  long rem = i % ((long)L * E);
  int e = (int)(rem % E);
  int l = (int)(rem / E);
  float base = xe[i] + pos[rem];
  if (e < 64) {
    int j = e >> 1;
    float f = (float)l * powf(10000.f, -(float)j / 32.f);
    float cs = cosf(f), sn = sinf(f);
    float partner = (e & 1) ? (xe[i - 1] + pos[rem - 1]) : (xe[i + 1] + pos[rem + 1]);
    out[i] = (e & 1) ? (base * cs + partner * sn) : (base * cs - partner * sn);
  } else {
    out[i] = base;
  }
}

__global__ void layernorm_rows_k(const float* __restrict__ x, const float* __restrict__ g,
                                 const float* __restrict__ bta, float* __restrict__ out,
                                 int D) {
  int row = blockIdx.x;
  int lane = threadIdx.x;
  const float* xr = x + (long)row * D;
  float s = 0.f, s2 = 0.f;
  for (int i = lane; i < D; i += 32) { float v = xr[i]; s += v; s2 += v * v; }
  for (int off = 16; off >= 1; off >>= 1) { s += __shfl_xor(s, off); s2 += __shfl_xor(s2, off); }
  float mean = s / D;
  float var = s2 / D - mean * mean;
  float inv = rsqrtf(var + 1e-5f);
  float* outr = out + (long)row * D;
  for (int i = lane; i < D; i += 32) outr[i] = (xr[i] - mean) * inv * g[i] + bta[i];
}

__global__ void softmax_rows_k(float* __restrict__ x, int D) {
  int row = blockIdx.x;
  int lane = threadIdx.x;
  float* xr = x + (long)row * D;
  float mx = -1e30f;
  for (int i = lane; i < D; i += 32) mx = fmaxf(mx, xr[i]);
  for (int off = 16; off >= 1; off >>= 1) mx = fmaxf(mx, __shfl_xor(mx, off));
  float s = 0.f;
  for (int i = lane; i < D; i += 32) { float e = __expf(xr[i] - mx); xr[i] = e; s += e; }
  for (int off = 16; off >= 1; off >>= 1) s += __shfl_xor(s, off);
  float invs = 1.f / s;
  for (int i = lane; i < D; i += 32) xr[i] *= invs;
}

__global__ void transpose01_k(const float* __restrict__ in, float* __restrict__ out,
                              int Bc, int D1, int D2) {
  long total = (long)Bc * D1 * D2;
  long i = blockIdx.x * (long)blockDim.x + threadIdx.x;
  if (i >= total) return;
  int j = (int)(i % D2); long t = i / D2;
  int ii = (int)(t % D1); int b = (int)(t / D1);
  out[((long)b * D2 + j) * D1 + ii] = in[i];
}

__global__ void reverse01_k(const float* __restrict__ in, float* __restrict__ out,
                            int Bc, int L, int D) {
  long total = (long)Bc * L * D;
  long i = blockIdx.x * (long)blockDim.x + threadIdx.x;
  if (i >= total) return;
  int d = (int)(i % D); long t = i / D;
  int l = (int)(t % L); int b = (int)(t / L);
  out[i] = in[((long)b * L + (L - 1 - l)) * D + d];
}

// causal depthwise conv1d (D_CONV=4, left pad 3) + bias + silu.  xh lives in xz
// at stride 2*di.
__global__ void conv1d_silu_k(const float* __restrict__ xz, const float* __restrict__ w,
                              const float* __restrict__ bias, float* __restrict__ u,
                              int Bc, int L, int di) {
  long total = (long)Bc * L * di;
  long i = blockIdx.x * (long)blockDim.x + threadIdx.x;
  if (i >= total) return;
  int d = (int)(i % di); long t = i / di;
  int l = (int)(t % L); int b = (int)(t / L);
  int tdi = 2 * di;
  float s = bias[d];
  for (int j = 0; j < 4; ++j) {
    int ls = l - 3 + j;
    if (ls >= 0) s += w[d * 4 + j] * xz[((long)b * L + ls) * tdi + d];
  }
  u[i] = s / (1.f + __expf(-s));
}

// selective scan: one block per batch, one thread per channel, 16 states in regs,
// per-step B/C (16+16) staged through LDS.
__global__ void mamba_scan_k(const float* __restrict__ dt, const float* __restrict__ xdbl,
                             const float* __restrict__ u, const float* __restrict__ A_log,
                             const float* __restrict__ Dp, float* __restrict__ y,
                             int L, int di, int nx, int dtr) {
  int b = blockIdx.x;
  int d = threadIdx.x;
  __shared__ float sBC[32];
  float Aa[16], h[16];
#pragma unroll
  for (int s = 0; s < 16; ++s) { Aa[s] = -__expf(A_log[d * 16 + s]); h[s] = 0.f; }
  float Dv = Dp[d];
  for (int l = 0; l < L; ++l) {
    long base = (long)b * L + l;
    if (d < 32) sBC[d] = xdbl[base * nx + dtr + d];
    __syncthreads();
    float dtv = dt[base * di + d];
    float uv = u[base * di + d];
    float accv = 0.f;
#pragma unroll
    for (int s = 0; s < 16; ++s) {
      h[s] = __expf(dtv * Aa[s]) * h[s] + dtv * sBC[s] * uv;
      accv += h[s] * sBC[16 + s];
    }
    y[base * di + d] = accv + uv * Dv;
    __syncthreads();
  }
}

__global__ void mul_silu_k(const float* __restrict__ y, const float* __restrict__ xz,
                           float* __restrict__ ym, int Bc, int L, int di) {
  long total = (long)Bc * L * di;
  long i = blockIdx.x * (long)blockDim.x + threadIdx.x;
  if (i >= total) return;
  int d = (int)(i % di); long t = i / di;
  int l = (int)(t % L); int b = (int)(t / L);
  float z = xz[((long)b * L + l) * (2 * di) + di + d];
  ym[i] = y[i] * (z / (1.f + __expf(-z)));
}

// orial[b, c, p] from spatial mamba outputs (B,128,256), with reverse on msb.
__global__ void orial_combine_k(const float* __restrict__ ms, const float* __restrict__ msb,
                                float* __restrict__ orial, int Bc) {
  long total = (long)Bc * 128 * 256;
  long i = blockIdx.x * (long)blockDim.x + threadIdx.x;
  if (i >= total) return;
  long rem = i % (128 * 256);
  int b = (int)(i / (128 * 256));
  int lrow = (int)(rem % 128);
  int dcol = (int)(rem / 128);
  orial[i] = ms[((long)b * 128 + lrow) * 256 + dcol] +
             msb[((long)b * 128 + (127 - lrow)) * 256 + dcol];
}

// per-(b,c): cfc_w[:,0]*mean + cfc_w[:,1]*std(ddof=1) + cfc_b
__global__ void se_stats_k(const float* __restrict__ orial, const float* __restrict__ cfc_w,
                           const float* __restrict__ cfc_b, float* __restrict__ out,
                           int Bc, int E, int HWc) {
  int total = Bc * E;
  int i = blockIdx.x * blockDim.x + threadIdx.x;
  if (i >= total) return;
  int c = i % E;
  const float* p = orial + (long)i * HWc;
  float s = 0.f;
  for (int k = 0; k < HWc; ++k) s += p[k];
  float mean = s / HWc;
  float s2 = 0.f;
  for (int k = 0; k < HWc; ++k) { float d = p[k] - mean; s2 += d * d; }
  float sd = sqrtf(s2 / (HWc - 1));
  out[i] = cfc_w[c * 2] * mean + cfc_w[c * 2 + 1] * sd + cfc_b[c];
}

// out[b,c,l] = xm[b, (rev? L-1-l : l), c] * score[b,c]
__global__ void scale_transpose_k(const float* __restrict__ xm, const float* __restrict__ score,
                                  float* __restrict__ out, int Bc, int L, int E, int rev) {
  long total = (long)Bc * E * L;
  long i = blockIdx.x * (long)blockDim.x + threadIdx.x;
  if (i >= total) return;
  int l = (int)(i % L); long t = i / L;
  int e = (int)(t % E); int b = (int)(t / E);
  int ls = rev ? (L - 1 - l) : l;
  out[i] = xm[((long)b * L + ls) * E + e] * score[b * E + e];
}

__global__ void pool_mean_k(const float* __restrict__ x, float* __restrict__ out,
                            int Bc, int L, int E) {
  int total = Bc * E;
  int i = blockIdx.x * blockDim.x + threadIdx.x;
  if (i >= total) return;
  int e = i % E; int b = i / E;
  float s = 0.f;
  for (int l = 0; l < L; ++l) s += x[((long)b * L + l) * E + e];
  out[i] = s / L;
}

__global__ void fuse_pool_k(const float* __restrict__ mt, const float* __restrict__ mc,
                            const float* __restrict__ srgb, const float* __restrict__ sdep,
                            const float* __restrict__ alpha, float* __restrict__ pr,
                            float* __restrict__ pd, float* __restrict__ pf, int n) {
  int i = blockIdx.x * blockDim.x + threadIdx.x;
  if (i >= n) return;
  float a = alpha[0];
  float r = mt[i] * srgb[i];
  float d = mc[i] * sdep[i];
  pr[i] = r; pd[i] = d; pf[i] = a * r + (1.f - a) * d;
}

// ---------------- host ----------------
extern "C" void kernel_launch(void* const* d_in, const int* in_sizes, int n_in,
                              void* d_out, int out_size, void* d_ws, size_t ws_size,
                              hipStream_t stream) {
  (void)in_sizes; (void)n_in; (void)out_size; (void)ws_size;
  auto IN = [&](int i) { return (const float*)d_in[i]; };

  // input index map (setup_inputs insertion order, depth-first)
  const int IX = 0, ICONV1 = 1, IBN2 = 2, IBN3 = 6, IBN4 = 10;
  const int IC2G = 14, IC2P = 15, IC3G = 16, IC3P = 17, ICONV4 = 18;
  const int IPATCHW = 19, IPATCHB = 20, IPOS = 21;
  const int ILN1G = 22, ILN1B = 23, ILNG = 24, ILNB = 25;
  const int IDCONV = 26, IWCONV = 27;
  const int IM0 = 28, IMB0 = 37, IMS0 = 46, IMSB0 = 55;
  const int ISE = 64;
  const int IC1D0W = 68, IC1D0B = 69, IC1D1W = 70, IC1D1B = 71, IC1D2W = 72, IC1D2B = 73;
  const int IGCN = 74;
  const int IFC1W = 78, IFC1B = 79, IFC2W = 80, IFC2B = 81;
  const int ITIM = 82;
  const int ICLS0 = 91, ICLS1 = 95, ICLS2 = 99;

  const size_t NBLE = (size_t)Bn * HWn * Emb;     // 1,048,576
  const size_t NBCL = (size_t)Bn * Cch * HWn;     // 1,638,400
  const size_t N8   = (size_t)Bn * HWn * HWn;     // 2,097,152

  float* wsf = (float*)d_ws;
  size_t off = 0;
  auto alloc = [&](size_t n) { float* p = wsf + off; off += n; return p; };

  float* m_xz   = alloc(N8);        // mamba xz (also x_cat later)
  float* w_att  = alloc(N8);        // GCN attention
  float* w_h1   = alloc(N8);        // MLP hidden
  float* w_xr   = alloc(NBCL);      // (B,C,HW)
  float* w_t    = alloc(NBLE);      // bn-relu temp / orial / LN temp
  float* w_dw   = alloc(NBLE);      // dw conv temp / s1
  float* w_pw   = alloc(NBLE);      // pw conv temp / s2
  float* w_x1   = alloc(NBLE);      // x0/x1/x2 / x_t
  float* w_xe   = alloc(NBLE);      // patch embed (long lived)
  float* w_x3   = alloc(NBLE);      // post-LN / x0n
  float* w_bel  = alloc(NBLE);      // (B,E,L) temp / g1
  float* w_spa  = alloc(NBLE);      // x_mm_spa / g2
  float* w_xmm  = alloc(NBLE);      // x_mm / x_g
  float* w_rev  = alloc(NBLE);      // reversed input / x_c
  float* m_u    = alloc(NBLE);
  float* m_dt   = alloc(NBLE);
  float* m_y    = alloc(NBLE);
  float* m_ym   = alloc(NBLE);
  float* o_m    = alloc(NBLE);
  float* o_mb   = alloc(NBLE);
  float* o_ms   = alloc(NBLE);
  float* o_msb  = alloc(NBLE);
  float* m_xdbl = alloc((size_t)Bn * HWn * 48);
  float* sm     = alloc(65536);
  float* w_cat  = m_xz;             // alias: xz dead after mamba section
  float* w_orial = w_t;
  float* sm_mt = sm, *sm_mc = sm + 4096, *sm_mid = sm + 8192, *sm_srgb = sm + 12288,
       *sm_sdep = sm + 16384, *sm_pr = sm + 20480, *sm_pd = sm + 24576, *sm_pf = sm + 28672,
       *sm_seo = sm + 32768, *sm_score = sm + 36864, *sm_ln = sm + 40960;

  auto g1d = [](long n) { return dim3((unsigned)((n + 255) / 256)); };
  auto gemm = [&](const float* A, const float* Bp, const float* bias, float* Cp,
                  int M, int N, int K, int lda, int ldb, int ldc,
                  long sA, long sB, long sC, int batch, int tA, int tB,
                  float alpha, int act) {
    dim3 grid((N + 31) / 32, (M + 31) / 32, batch);
    dim3 blk(32);
    if (!tA && !tB)
      gemm_wmma_f32<0, 0><<<grid, blk, 0, stream>>>(A, Bp, bias, Cp, M, N, K, lda, ldb, ldc,
                                                    sA, sB, sC, alpha, act);
    else if (!tA && tB)
      gemm_wmma_f32<0, 1><<<grid, blk, 0, stream>>>(A, Bp, bias, Cp, M, N, K, lda, ldb, ldc,
                                                    sA, sB, sC, alpha, act);
    else if (tA && !tB)
      gemm_wmma_f32<1, 0><<<grid, blk, 0, stream>>>(A, Bp, bias, Cp, M, N, K, lda, ldb, ldc,
                                                    sA, sB, sC, alpha, act);
    else
      gemm_wmma_f32<1, 1><<<grid, blk, 0, stream>>>(A, Bp, bias, Cp, M, N, K, lda, ldb, ldc,
                                                    sA, sB, sC, alpha, act);
  };

  const float* x = IN(IX);

  // ---- conv stem ----
  // x0 = conv1(xi): per b, (E x C) @ (C x HW)
  gemm(IN(ICONV1), x, nullptr, w_x1, Emb, HWn, Cch, Cch, HWn, HWn,
       0, (long)Cch * HWn, (long)Emb * HWn, Bn, 0, 0, 1.f, 0);
  auto bnrelu = [&](const float* src, int bnIdx, float* dst) {
    bn_relu_k<<<g1d(NBLE), 256, 0, stream>>>(src, IN(bnIdx), IN(bnIdx + 1), IN(bnIdx + 2),
                                             IN(bnIdx + 3), dst, (long)NBLE, Emb, HWn);
  };
  auto hsc = [&](const float* gw, const float* pw) {  // in: w_t, out: w_x1
    dwconv3x3_k<<<g1d(NBLE), 256, 0, stream>>>(w_t, gw, w_dw, Bn, Emb, Himg);
    gemm(pw, w_t, nullptr, w_pw, Emb, HWn, Emb, Emb, HWn, HWn,
         0, (long)Emb * HWn, (long)Emb * HWn, Bn, 0, 0, 1.f, 0);
    add2_k<<<g1d(NBLE), 256, 0, stream>>>(w_x1, w_dw, w_pw, (long)NBLE);
  };
  bnrelu(w_x1, IBN2, w_t); hsc(IN(IC2G), IN(IC2P));                       // x1
  bnrelu(w_x1, IBN3, w_t); hsc(IN(IC3G), IN(IC3P));                       // x2
  bnrelu(w_x1, IBN4, w_t);
  gemm(IN(ICONV4), w_t, nullptr, w_xr, Cch, HWn, Emb, Emb, HWn, HWn,
       0, (long)Emb * HWn, (long)Cch * HWn, Bn, 0, 0, 1.f, 0);
  add2_k<<<g1d(NBCL), 256, 0, stream>>>(w_xr, w_xr, x, (long)NBCL);       // xr

  // ---- patch embed: xe[b,l,e] = sum_c xr[b,c,l]*patch_w[e,c] + patch_b ----
  gemm(w_xr, IN(IPATCHW), IN(IPATCHB), w_xe, HWn, Emb, Cch, HWn, Cch, Emb,
       (long)Cch * HWn, 0, (long)HWn * Emb, Bn, 1, 1, 1.f, 0);

  // ---- rope(xe+pos) then LN ----
  rope_pos_k<<<g1d(NBLE), 256, 0, stream>>>(w_xe, IN(IPOS), w_t, Bn, HWn, Emb);
  layernorm_rows_k<<<Bn * HWn, 32, 0, stream>>>(w_t, IN(ILN1G), IN(ILN1B), w_x3, Emb);

  // ---- dconv + wconv ----
  transpose01_k<<<g1d(NBLE), 256, 0, stream>>>(w_x3, w_bel, Bn, HWn, Emb);  // (B,E,L)
  dwconv3x3_k<<<g1d(NBLE), 256, 0, stream>>>(w_bel, IN(IDCONV), w_dw, Bn, Emb, Himg);
  gemm(IN(IWCONV), w_dw, nullptr, w_spa, Emb, HWn, Emb, Emb, HWn, HWn,
       0, (long)Emb * HWn, (long)Emb * HWn, Bn, 0, 0, 1.f, 0);             // x_mm_spa (B,E,HW)
  transpose01_k<<<g1d(NBLE), 256, 0, stream>>>(w_spa, w_xmm, Bn, Emb, HWn); // x_mm (B,HW,E)

  // ---- mamba blocks ----
  auto run_mamba = [&](const float* X, int L, int dm, int pi, float* out) {
    int di = dm, dtr = (dm + 15) / 16, nx = dtr + 32;
    long sX = (long)L * dm;
    gemm(X, IN(pi + 0), nullptr, m_xz, L, 2 * di, dm, dm, dm, 2 * di,
         sX, 0, (long)L * 2 * di, Bn, 0, 1, 1.f, 0);
    conv1d_silu_k<<<g1d((long)Bn * L * di), 256, 0, stream>>>(m_xz, IN(pi + 1), IN(pi + 2),
                                                              m_u, Bn, L, di);
    gemm(m_u, IN(pi + 3), nullptr, m_xdbl, L, nx, di, di, di, nx,
         (long)L * di, 0, (long)L * nx, Bn, 0, 1, 1.f, 0);
    gemm(m_xdbl, IN(pi + 4), IN(pi + 5), m_dt, L, di, dtr, nx, dtr, di,
         (long)L * nx, 0, (long)L * di, Bn, 0, 1, 1.f, 5);                 // softplus
    mamba_scan_k<<<Bn, di, 0, stream>>>(m_dt, m_xdbl, m_u, IN(pi + 6), IN(pi + 7), m_y,
                                        L, di, nx, dtr);
    mul_silu_k<<<g1d((long)Bn * L * di), 256, 0, stream>>>(m_y, m_xz, m_ym, Bn, L, di);
    gemm(m_ym, IN(pi + 8), nullptr, out, L, dm, di, di, di, dm,
         (long)L * di, 0, (long)L * dm, Bn, 0, 1, 1.f, 0);
  };
  run_mamba(w_xmm, HWn, Emb, IM0, o_m);
  reverse01_k<<<g1d(NBLE), 256, 0, stream>>>(w_xmm, w_rev, Bn, HWn, Emb);
  run_mamba(w_rev, HWn, Emb, IMB0, o_mb);
  run_mamba(w_spa, Emb, HWn, IMS0, o_ms);
  reverse01_k<<<g1d(NBLE), 256, 0, stream>>>(w_spa, w_rev, Bn, Emb, HWn);
  run_mamba(w_rev, Emb, HWn, IMSB0, o_msb);

  // ---- SE gate + conv1d mixing ----
  orial_combine_k<<<g1d(NBLE), 256, 0, stream>>>(o_ms, o_msb, w_orial, Bn);
  se_stats_k<<<g1d(Bn * Emb), 256, 0, stream>>>(w_orial, IN(ISE), IN(ISE + 1), sm_seo,
                                                Bn, Emb, HWn);
  gemm(sm_seo, IN(ISE + 2), IN(ISE + 3), sm_score, Bn, Emb, Emb, Emb, Emb, Emb,
       0, 0, 0, 1, 0, 1, 1.f, 2);                                          // sigmoid
  scale_transpose_k<<<g1d(NBLE), 256, 0, stream>>>(o_m, sm_score, w_dw, Bn, HWn, Emb, 0);
  scale_transpose_k<<<g1d(NBLE), 256, 0, stream>>>(o_mb, sm_score, w_pw, Bn, HWn, Emb, 1);
  gemm(w_dw, IN(IC1D0W), IN(IC1D0B), w_cat, HWn, Emb, Emb, HWn, Emb, 2 * Emb,
       (long)Emb * HWn, 0, (long)HWn * 2 * Emb, Bn, 1, 1, 1.f, 0);
  gemm(w_pw, IN(IC1D1W), IN(IC1D1B), w_cat + Emb, HWn, Emb, Emb, HWn, Emb, 2 * Emb,
       (long)Emb * HWn, 0, (long)HWn * 2 * Emb, Bn, 1, 1, 1.f, 0);
  gemm(w_cat, IN(IC1D2W), IN(IC1D2B), w_x1, HWn, Emb, 2 * Emb, 2 * Emb, 2 * Emb, Emb,
       (long)HWn * 2 * Emb, 0, (long)HWn * Emb, Bn, 0, 1, 1.f, 0);         // x_t -> w_x1

  // ---- GCN branch ----
  layernorm_rows_k<<<Bn * HWn, 32, 0, stream>>>(w_xe, IN(ILNG), IN(ILNB), w_x3, Emb);
  gemm(w_x3, w_x3, nullptr, w_att, HWn, HWn, Emb, Emb, Emb, HWn,
       (long)HWn * Emb, (long)HWn * Emb, (long)HWn * HWn, Bn, 0, 1,
       0.08838834765f, 0);                                                 // X@X^T / sqrt(d)
  softmax_rows_k<<<Bn * HWn, 32, 0, stream>>>(w_att, HWn);
  gemm(w_x3, IN(IGCN), nullptr, w_bel, HWn, Emb, Emb, Emb, Emb, Emb,
       (long)HWn * Emb, 0, (long)HWn * Emb, Bn, 0, 0, 1.f, 0);             // x@w1
  gemm(w_att, w_bel, IN(IGCN + 1), w_spa, HWn, Emb, HWn, HWn, Emb, Emb,
       (long)HWn * HWn, (long)HWn * Emb, (long)HWn * Emb, Bn, 0, 0, 1.f, 1); // relu
  gemm(w_spa, IN(IGCN + 2), nullptr, w_bel, HWn, Emb, Emb, Emb, Emb, Emb,
       (long)HWn * Emb, 0, (long)HWn * Emb, Bn, 0, 0, 1.f, 0);             // h@w2
  gemm(w_att, w_bel, IN(IGCN + 3), w_spa, HWn, Emb, HWn, HWn, Emb, Emb,
       (long)HWn * HWn, (long)HWn * Emb, (long)HWn * Emb, Bn, 0, 0, 1.f, 0);
  add2_k<<<g1d(NBLE), 256, 0, stream>>>(w_xmm, w_spa, w_xe, (long)NBLE);   // x_g -> w_xmm

  // ---- MLP ----
  layernorm_rows_k<<<Bn * HWn, 32, 0, stream>>>(w_xmm, IN(ILNG), IN(ILNB), w_t, Emb);
  gemm(w_t, IN(IFC1W), IN(IFC1B), w_h1, HWn, 2 * Emb, Emb, Emb, Emb, 2 * Emb,
       (long)HWn * Emb, 0, (long)HWn * 2 * Emb, Bn, 0, 1, 1.f, 4);         // gelu
  gemm(w_h1, IN(IFC2W), IN(IFC2B), w_rev, HWn, Emb, 2 * Emb, 2 * Emb, 2 * Emb, Emb,
       (long)HWn * 2 * Emb, 0, (long)HWn * Emb, Bn, 0, 1, 1.f, 0);
  add2_k<<<g1d(NBLE), 256, 0, stream>>>(w_rev, w_rev, w_xmm, (long)NBLE);  // x_c -> w_rev

  // ---- SE2D fusion (pooled form) + classifiers ----
  pool_mean_k<<<g1d(Bn * Emb), 256, 0, stream>>>(w_x1, sm_mt, Bn, HWn, Emb);
  pool_mean_k<<<g1d(Bn * Emb), 256, 0, stream>>>(w_rev, sm_mc, Bn, HWn, Emb);
  gemm(sm_mt, IN(ITIM + 0), IN(ITIM + 1), sm_mid, Bn, 8, Emb, Emb, Emb, 8,
       0, 0, 0, 1, 0, 1, 1.f, 1);
  gemm(sm_mid, IN(ITIM + 2), IN(ITIM + 3), sm_srgb, Bn, Emb, 8, 8, 8, Emb,
       0, 0, 0, 1, 0, 1, 1.f, 2);
  gemm(sm_mc, IN(ITIM + 4), IN(ITIM + 5), sm_mid, Bn, 8, Emb, Emb, Emb, 8,
       0, 0, 0, 1, 0, 1, 1.f, 1);
  gemm(sm_mid, IN(ITIM + 6), IN(ITIM + 7), sm_sdep, Bn, Emb, 8, 8, 8, Emb,
       0, 0, 0, 1, 0, 1, 1.f, 2);
  fuse_pool_k<<<g1d(Bn * Emb), 256, 0, stream>>>(sm_mt, sm_mc, sm_srgb, sm_sdep,
                                                 IN(ITIM + 8), sm_pr, sm_pd, sm_pf,
                                                 Bn * Emb);
  float* outp = (float*)d_out;
  auto classify = [&](const float* src, int ci, float* dst) {
    layernorm_rows_k<<<Bn, 32, 0, stream>>>(src, IN(ci), IN(ci + 1), sm_ln, Emb);
    gemm(sm_ln, IN(ci + 2), IN(ci + 3), dst, Bn, NCls, Emb, Emb, Emb, NCls,
         0, 0, 0, 1, 0, 1, 1.f, 0);
  };
  classify(sm_pf, ICLS0, outp);
  classify(sm_pr, ICLS1, outp + Bn * NCls);
  classify(sm_pd, ICLS2, outp + 2 * Bn * NCls);
}